// CreateOverlappingWindows_84963043050043
// MI455X (gfx1250) — compile-verified
//
#include <hip/hip_runtime.h>
#include <stdint.h>

// Problem constants (from reference): x (64, 2000, 26) f32 -> out (128000, 494) f32
#define NCTX        9
#define WIN         19            // 2*NCTX+1
#define CIN         26
#define TT          2000
#define BB          64
#define ROWLEN      (WIN * CIN)   // 494 floats per output row
#define T_TILE      125           // 2000 / 125 = 16 exact tiles
#define TILES_PER_B (TT / T_TILE) // 16
#define HALO_FRAMES (T_TILE + 2 * NCTX)   // 143
#define NELEM       (HALO_FRAMES * CIN)   // 3718 floats in LDS tile
#define NPAIR       (NELEM / 2)           // 1859 float2 pairs

__global__ __launch_bounds__(256)
void CreateOverlappingWindows_win_kernel(const float* __restrict__ x,
                                         float* __restrict__ out) {
    __shared__ float smem[NELEM];

    const int tid  = threadIdx.x;
    const int tile = blockIdx.x % TILES_PER_B;
    const int b    = blockIdx.x / TILES_PER_B;
    const int t0   = tile * T_TILE;

    // Tile source: frames [t0-9, t0+T_TILE+9) of batch b (contiguous).
    // Pointer may sit before the batch start; only dereferenced where valid.
    const float* gsrc = x + ((long)b * TT + (t0 - NCTX)) * CIN;

    // Low 32 bits of a flat pointer into LDS == wave-relative LDS byte offset.
    const uint32_t lds_base = (uint32_t)(uintptr_t)(&smem[0]);

    // ---- Phase 1: stage halo tile into LDS via async global->LDS DMA path ----
    // Pairs (i, i+1) always lie in one frame (CIN even), so validity is per-pair.
    for (int p = tid; p < NPAIR; p += 256) {
        const int i = 2 * p;
        const int f = t0 - NCTX + (i / CIN);          // source frame index
        const uint32_t lds_off = lds_base + (uint32_t)(i * 4);
        if (f >= 0 && f < TT) {
            const uint64_t ga = (uint64_t)(uintptr_t)(gsrc + i);   // 8B aligned
            asm volatile("global_load_async_to_lds_b64 %0, %1, off"
                         :: "v"(lds_off), "v"(ga) : "memory");
        } else {
            // zero-pad halo (DS store; disjoint from async-load targets)
            float2 z; z.x = 0.0f; z.y = 0.0f;
            *(float2*)(&smem[i]) = z;
        }
    }
    asm volatile("s_wait_asynccnt 0x0" ::: "memory");  // my async loads landed
    __syncthreads();                                   // everyone's landed

    // ---- Phase 2: stream output rows LDS->global via async stores ----
    // Output row (b, t0+r) == LDS floats [r*26, r*26+494): contiguous slide.
    if (tid < ROWLEN / 2) {                            // 247 active lanes
        float*   grow = out + ((long)b * TT + t0) * ROWLEN + 2 * tid;
        uint32_t lrow = lds_base + (uint32_t)(2 * tid * 4);
        for (int r = 0; r < T_TILE; ++r) {
            const uint64_t ga = (uint64_t)(uintptr_t)grow;         // 8B aligned
            asm volatile("global_store_async_from_lds_b64 %0, %1, off"
                         :: "v"(ga), "v"(lrow) : "memory");
            grow += ROWLEN;        // next output row
            lrow += CIN * 4;       // slide window by one frame in LDS
        }
    }
    // Drain async stores before wave teardown (endpgm also implies wait-idle).
    asm volatile("s_wait_asynccnt 0x0" ::: "memory");
}

extern "C" void kernel_launch(void* const* d_in, const int* in_sizes, int n_in,
                              void* d_out, int out_size, void* d_ws, size_t ws_size,
                              hipStream_t stream) {
    const float* x   = (const float*)d_in[0];
    float*       out = (float*)d_out;
    (void)in_sizes; (void)n_in; (void)out_size; (void)d_ws; (void)ws_size;

    dim3 grid(BB * TILES_PER_B);   // 64 * 16 = 1024 blocks
    dim3 block(256);               // 8 wave32
    CreateOverlappingWindows_win_kernel<<<grid, block, 0, stream>>>(x, out);
}